// MixedLoss_7524782702792
// MI455X (gfx1250) — compile-verified
//
#include <hip/hip_runtime.h>
#include <hip/hip_bf16.h>

typedef float v2f __attribute__((ext_vector_type(2)));
typedef float v8f __attribute__((ext_vector_type(8)));

#define IMG 384
#define WIN 378
#define CROP 3
#define SHIFTS 7
#define NSHIFT 49
#define BATCH 64
#define NTOT (BATCH * IMG * IMG)                 /* 9437184 */
#define NWIN 9144576.0f                          /* 64*378*378 */
#define THRESHV (-0.995f)
#define ROWCHUNKS (IMG / 4)                      /* 96 x b128 per row */

/* workspace layout (floats) */
#define WS_MSE 0
#define WS_SP  1
#define WS_SP2 2
#define WS_X   16
#define WS_SU  80
#define WS_SU2 144
#define WS_C   208
#define WS_N   512

__global__ void k_init(float* ws) {
    int t = threadIdx.x;
    if (t < WS_N) ws[t] = 0.0f;
}

/* ---------------- Kernel 1: MSE over full tensor + SP/SP2 over the crop ---- */
__global__ void k_mse(const float* __restrict__ pred,
                      const float* __restrict__ blended,
                      float* __restrict__ ws) {
    int tid = threadIdx.x;
    size_t g = (size_t)blockIdx.x * blockDim.x + tid;
    size_t stride = (size_t)gridDim.x * blockDim.x;
    float mse = 0.0f, sp = 0.0f, sp2 = 0.0f;
    for (size_t idx = g; idx < (size_t)NTOT; idx += stride) {
        float p = pred[idx];
        float d = p - blended[idx];
        mse = fmaf(d, d, mse);
        int x = (int)(idx % IMG);
        int y = (int)((idx / IMG) % IMG);
        if (x >= CROP && x < CROP + WIN && y >= CROP && y < CROP + WIN) {
            sp += p;
            sp2 = fmaf(p, p, sp2);
        }
    }
    /* wave reduce then block reduce */
    #pragma unroll
    for (int off = 16; off > 0; off >>= 1) {
        mse += __shfl_down(mse, off, 32);
        sp  += __shfl_down(sp,  off, 32);
        sp2 += __shfl_down(sp2, off, 32);
    }
    __shared__ float r[24];
    int lane = tid & 31, wave = tid >> 5;
    if (lane == 0) { r[wave] = mse; r[8 + wave] = sp; r[16 + wave] = sp2; }
    __syncthreads();
    if (tid == 0) {
        float m = 0.0f, a = 0.0f, b = 0.0f;
        #pragma unroll
        for (int w = 0; w < 8; ++w) { m += r[w]; a += r[8 + w]; b += r[16 + w]; }
        atomicAdd(&ws[WS_MSE], m);
        atomicAdd(&ws[WS_SP],  a);
        atomicAdd(&ws[WS_SP2], b);
    }
}

/* ------- Kernel 2: separable window sums SU / SU2 / C for all 49 shifts ---- */
/* one block per image; 384 threads, thread x owns column x (coalesced rows)  */
__global__ void k_winstats(const float* __restrict__ U, float* __restrict__ ws) {
    __shared__ float s[21 * IMG];
    int b = blockIdx.x;
    int x = threadIdx.x;
    const float* Ub = U + (size_t)b * IMG * IMG;
    float wv[7], wv2[7], wc[7];
    #pragma unroll
    for (int u = 0; u < 7; ++u) { wv[u] = 0.0f; wv2[u] = 0.0f; wc[u] = 0.0f; }
    for (int y = 0; y < IMG; ++y) {
        float val = Ub[(size_t)y * IMG + x];
        float v2 = val * val;
        float c = (val < THRESHV) ? 1.0f : 0.0f;
        #pragma unroll
        for (int u = 0; u < 7; ++u) {
            if (y >= u && y <= u + WIN - 1) { wv[u] += val; wv2[u] += v2; wc[u] += c; }
        }
    }
    #pragma unroll
    for (int u = 0; u < 7; ++u) {
        s[(0 * 7 + u) * IMG + x] = wv[u];
        s[(1 * 7 + u) * IMG + x] = wv2[u];
        s[(2 * 7 + u) * IMG + x] = wc[u];
    }
    __syncthreads();
    if (threadIdx.x < 21) {
        int stat = threadIdx.x / 7;
        int u = threadIdx.x % 7;
        const float* row = &s[threadIdx.x * IMG];
        float W[7];
        #pragma unroll
        for (int v = 0; v < 7; ++v) W[v] = 0.0f;
        for (int xx = 0; xx < IMG; ++xx) {
            float val = row[xx];
            #pragma unroll
            for (int v = 0; v < 7; ++v)
                if (xx >= v && xx <= v + WIN - 1) W[v] += val;
        }
        float* dst = ws + (stat == 0 ? WS_SU : (stat == 1 ? WS_SU2 : WS_C));
        #pragma unroll
        for (int v = 0; v < 7; ++v) atomicAdd(&dst[u * SHIFTS + v], W[v]);
    }
}

/* --- Kernel 3: 7x7 cross-correlation X(u,v) = sum U[u+i,v+j]*P[3+i,3+j] ---- */
/* block = (16-row band, image). Async b128 global->LDS staging (ASYNCcnt);   */
/* register-blocked FMA; per-wave shuffle reduce; cross-wave reduce via       */
/* V_WMMA_F32_16X16X4_F32 (ones x B).                                          */
__global__ void k_xcorr(const float* __restrict__ U,
                        const float* __restrict__ P,
                        float* __restrict__ ws) {
    __shared__ __align__(16) float sU[22 * IMG];  /* rows i0 .. i0+R+5        */
    __shared__ __align__(16) float sP[16 * IMG];  /* rows 3+i0 .. 3+i0+R-1    */
    __shared__ float sRed[8 * 64];                /* 8 wave-partials x 64     */

    int tid = threadIdx.x;
    int b = blockIdx.y;
    int i0 = blockIdx.x * 16;
    int R = (WIN - i0 < 16) ? (WIN - i0) : 16;
    const float* Ub = U + (size_t)b * IMG * IMG;
    const float* Pb = P + ((size_t)b * IMG + (CROP + i0)) * IMG;

    /* ---- async global->LDS staging, 16B per instruction, GVS addressing --- */
    unsigned ldsU0 = (unsigned)(uintptr_t)&sU[0];
    unsigned ldsP0 = (unsigned)(uintptr_t)&sP[0];
    int nU = (R + 6) * ROWCHUNKS;
    for (int t = tid; t < nU; t += blockDim.x) {
        unsigned ldsa = ldsU0 + (unsigned)t * 16u;
        unsigned goff = ((unsigned)i0 * ROWCHUNKS + (unsigned)t) * 16u;
        asm volatile("global_load_async_to_lds_b128 %0, %1, %2"
                     :: "v"(ldsa), "v"(goff), "s"(Ub) : "memory");
    }
    int nP = R * ROWCHUNKS;
    for (int t = tid; t < nP; t += blockDim.x) {
        unsigned ldsa = ldsP0 + (unsigned)t * 16u;
        unsigned goff = (unsigned)t * 16u;
        asm volatile("global_load_async_to_lds_b128 %0, %1, %2"
                     :: "v"(ldsa), "v"(goff), "s"(Pb) : "memory");
    }
    asm volatile("s_wait_asynccnt 0" ::: "memory");
    __syncthreads();

    float acc[NSHIFT];
    #pragma unroll
    for (int q = 0; q < NSHIFT; ++q) acc[q] = 0.0f;

    /* 63 strips of 6 output columns per SR row */
    int ntask = R * 63;
    for (int task = tid; task < ntask; task += blockDim.x) {
        int i = task / 63;
        int j0 = (task % 63) * 6;
        float pv[6];
        #pragma unroll
        for (int t = 0; t < 6; ++t) pv[t] = sP[i * IMG + CROP + j0 + t];
        #pragma unroll
        for (int u = 0; u < 7; ++u) {
            const float* ur = &sU[(i + u) * IMG + j0];
            float uv[12];
            #pragma unroll
            for (int t = 0; t < 12; ++t) uv[t] = ur[t];
            #pragma unroll
            for (int v = 0; v < 7; ++v) {
                #pragma unroll
                for (int t = 0; t < 6; ++t)
                    acc[u * 7 + v] = fmaf(uv[v + t], pv[t], acc[u * 7 + v]);
            }
        }
    }

    /* per-wave shuffle reduction -> sRed[wave][q] */
    int lane = tid & 31, wave = tid >> 5;
    #pragma unroll
    for (int q = 0; q < NSHIFT; ++q) {
        float v = acc[q];
        v += __shfl_down(v, 16, 32);
        v += __shfl_down(v, 8, 32);
        v += __shfl_down(v, 4, 32);
        v += __shfl_down(v, 2, 32);
        v += __shfl_down(v, 1, 32);
        if (lane == 0) sRed[wave * 64 + q] = v;
    }
    if (lane == 0)
        for (int q = NSHIFT; q < 64; ++q) sRed[wave * 64 + q] = 0.0f;
    __syncthreads();

    /* wave 0: column sums of the 8x64 matrix via WMMA with A = ones.
       D[m][n] = sum_k B[k][n] + C[m][n]; column sums are invariant to the
       exact k slot mapping as long as column == lane%16. */
    if (tid < 32) {
        int col = tid & 15;
        int kh = tid >> 4;               /* half-wave selects row pair */
        v2f ones; ones.x = 1.0f; ones.y = 1.0f;
        v8f d0 = {}, d1 = {}, d2 = {}, d3 = {};
        #pragma unroll
        for (int t = 0; t < 2; ++t) {    /* two 4-row tiles cover 8 rows */
            int r0 = 4 * t + 2 * kh;
            int r1 = r0 + 1;
            v2f b0, b1, b2, b3;
            b0.x = sRed[r0 * 64 +  0 + col]; b0.y = sRed[r1 * 64 +  0 + col];
            b1.x = sRed[r0 * 64 + 16 + col]; b1.y = sRed[r1 * 64 + 16 + col];
            b2.x = sRed[r0 * 64 + 32 + col]; b2.y = sRed[r1 * 64 + 32 + col];
            b3.x = sRed[r0 * 64 + 48 + col]; b3.y = sRed[r1 * 64 + 48 + col];
            d0 = __builtin_amdgcn_wmma_f32_16x16x4_f32(false, ones, false, b0, (short)0, d0, false, false);
            d1 = __builtin_amdgcn_wmma_f32_16x16x4_f32(false, ones, false, b1, (short)0, d1, false, false);
            d2 = __builtin_amdgcn_wmma_f32_16x16x4_f32(false, ones, false, b2, (short)0, d2, false, false);
            d3 = __builtin_amdgcn_wmma_f32_16x16x4_f32(false, ones, false, b3, (short)0, d3, false, false);
        }
        if (tid < 16) {  /* lanes 0-15: element 0 = row M=0 = column sum */
            atomicAdd(&ws[WS_X +  0 + col], d0[0]);
            atomicAdd(&ws[WS_X + 16 + col], d1[0]);
            atomicAdd(&ws[WS_X + 32 + col], d2[0]);
            if (col == 0) atomicAdd(&ws[WS_X + 48], d3[0]);
        }
    }
}

/* ---------------- Kernel 4: combine, argmin (first-min), write outputs ----- */
__global__ void k_final(const float* __restrict__ ws, float* __restrict__ out) {
    __shared__ float cm[NSHIFT];
    int s = threadIdx.x;
    if (s < NSHIFT) {
        float SP  = ws[WS_SP];
        float SP2 = ws[WS_SP2];
        float S1 = ws[WS_SU + s] - SP;
        float S2 = ws[WS_SU2 + s] - 2.0f * ws[WS_X + s] + SP2;
        float co = 1.0f / ws[WS_C + s];
        float bb = co * S1;
        cm[s] = co * (S2 - 2.0f * bb * S1 + NWIN * bb * bb);
    }
    __syncthreads();
    if (s == 0) {
        int best = 0;
        float bv = cm[0];
        for (int q = 1; q < NSHIFT; ++q)
            if (cm[q] < bv) { bv = cm[q]; best = q; }
        out[0] = ws[WS_MSE] / (float)NTOT;
        out[1] = bv;
        out[2] = (float)(best / SHIFTS);
        out[3] = (float)(best % SHIFTS);
    }
}

extern "C" void kernel_launch(void* const* d_in, const int* in_sizes, int n_in,
                              void* d_out, int out_size, void* d_ws, size_t ws_size,
                              hipStream_t stream) {
    const float* pred = (const float*)d_in[0];
    const float* unbl = (const float*)d_in[1];
    const float* blen = (const float*)d_in[2];
    float* ws = (float*)d_ws;
    float* out = (float*)d_out;

    k_init<<<1, 512, 0, stream>>>(ws);
    k_mse<<<1024, 256, 0, stream>>>(pred, blen, ws);
    k_winstats<<<BATCH, IMG, 0, stream>>>(unbl, ws);
    dim3 g3((WIN + 15) / 16, BATCH);
    k_xcorr<<<g3, 256, 0, stream>>>(unbl, pred, ws);
    k_final<<<1, 64, 0, stream>>>(ws, out);
}